// AngularBasis_34359738368388
// MI455X (gfx1250) — compile-verified
//
#include <hip/hip_runtime.h>
#include <cstdint>
#include <cstddef>

#define LMAX 10
#define BLK 256u

#if defined(__has_builtin)
#if __has_builtin(__builtin_amdgcn_tensor_store_from_lds)
#define HAVE_TDM 1
#endif
#endif

typedef unsigned int uint32x4 __attribute__((ext_vector_type(4)));
typedef int int32x8 __attribute__((ext_vector_type(8)));
typedef int int32x4 __attribute__((ext_vector_type(4)));

// ---------------- compile-time normalization table ----------------
// norm(l,m) = sqrt((2l+1)/(4*pi) * (l-m)!/(l+m)!)
// stored factor: m==0 -> norm ; m>0 -> sqrt(2)*(-1)^m*norm  (cancels CS phase)
constexpr double csqrt(double x) {
  double g = x > 1.0 ? x : 1.0;
  for (int i = 0; i < 128; ++i) g = 0.5 * (g + x / g);
  return g;
}

struct Norms { float f[LMAX + 1][LMAX + 1]; };

constexpr Norms makeNorms() {
  Norms t{};
  for (int l = 0; l <= LMAX; ++l) {
    for (int m = 0; m <= l; ++m) {
      double fr = 1.0;
      for (int k = l - m + 1; k <= l + m; ++k) fr /= (double)k;
      double nrm = csqrt((2.0 * l + 1.0) / (4.0 * 3.14159265358979323846) * fr);
      double v = (m == 0) ? nrm
                          : 1.4142135623730950488 * nrm * ((m & 1) ? -1.0 : 1.0);
      t.f[l][m] = (float)v;
    }
  }
  return t;
}

constexpr Norms kN = makeNorms();  // folds to inline literals in device code

// ---------------- kernel ----------------
// Block output tile staged in LDS laid out exactly like the global destination:
// region l starts at word 256*l*l, element (p, j) at p*(2l+1)+j.
// Odd word stride (2l+1) across lanes => conflict-free on 64 banks.
__global__ __launch_bounds__(BLK) void sph_harm_l10_kernel(
    const float* __restrict__ cos_theta, const float* __restrict__ phi,
    float* __restrict__ out, unsigned n) {
  __shared__ __align__(16) float tile[BLK * (LMAX + 1) * (LMAX + 1)];  // 121 KB
  __shared__ float s_ct[BLK];
  __shared__ float s_ph[BLK];

  const unsigned tid = threadIdx.x;
  const unsigned blockStart = blockIdx.x * BLK;
  const unsigned gi  = blockStart + tid;
  const unsigned idx = (gi < n) ? gi : (n - 1u);  // clamp tail for async load

  // ---- CDNA5 async global->LDS staging of this block's inputs (ASYNCcnt) ----
  {
    unsigned lds_ct = (unsigned)(uintptr_t)&s_ct[tid];  // low 32b of flat LDS addr == LDS offset
    unsigned lds_ph = (unsigned)(uintptr_t)&s_ph[tid];
    unsigned voff   = idx * 4u;                          // byte offset (GVS mode)
    unsigned long long bct = (unsigned long long)(uintptr_t)cos_theta;
    unsigned long long bph = (unsigned long long)(uintptr_t)phi;
    asm volatile("global_load_async_to_lds_b32 %0, %1, %2"
                 :: "v"(lds_ct), "v"(voff), "s"(bct) : "memory");
    asm volatile("global_load_async_to_lds_b32 %0, %1, %2"
                 :: "v"(lds_ph), "v"(voff), "s"(bph) : "memory");
    asm volatile("s_wait_asynccnt 0" ::: "memory");
  }

  const float x  = s_ct[tid];
  const float ph = s_ph[tid];

  // ---------------- compute phase (predicated; all lanes reach barrier) ----
  if (gi < n) {
    const float st = __builtin_sqrtf(__builtin_fmaxf(1.0f - x * x, 0.0f));
    float s1, c1;
    __sincosf(ph, &s1, &c1);

    float pmm = 1.0f;  // P_{m,m} (with Condon-Shortley phase)
    float cm  = 1.0f;  // cos(m*phi)
    float sm  = 0.0f;  // sin(m*phi)

#pragma unroll
    for (int m = 0; m <= LMAX; ++m) {
      float plm2 = pmm;  // P_{m,m}
      {
        const int l = m;
        float* rowp = &tile[BLK * (unsigned)(l * l) + tid * (unsigned)(2 * l + 1)];
        const float v = kN.f[l][m] * plm2;
        if (m == 0) {
          rowp[l] = v;
        } else {
          rowp[l + m] = v * cm;
          rowp[l - m] = v * sm;
        }
      }
      if (m < LMAX) {
        float plm1 = (float)(2 * m + 1) * x * pmm;  // P_{m+1,m}
        {
          const int l = m + 1;
          float* rowp = &tile[BLK * (unsigned)(l * l) + tid * (unsigned)(2 * l + 1)];
          const float v = kN.f[l][m] * plm1;
          if (m == 0) {
            rowp[l] = v;
          } else {
            rowp[l + m] = v * cm;
            rowp[l - m] = v * sm;
          }
        }
#pragma unroll
        for (int l = m + 2; l <= LMAX; ++l) {
          const float A = (float)(2 * l - 1) / (float)(l - m);  // compile-time literal
          const float B = (float)(l + m - 1) / (float)(l - m);
          const float pl = A * x * plm1 - B * plm2;
          plm2 = plm1;
          plm1 = pl;
          float* rowp = &tile[BLK * (unsigned)(l * l) + tid * (unsigned)(2 * l + 1)];
          const float v = kN.f[l][m] * pl;
          if (m == 0) {
            rowp[l] = v;
          } else {
            rowp[l + m] = v * cm;
            rowp[l - m] = v * sm;
          }
        }
      }
      pmm = -(float)(2 * m + 1) * st * pmm;  // P_{m+1,m+1}
      const float cn  = cm * c1 - sm * s1;   // cos((m+1)phi)
      const float snn = sm * c1 + cm * s1;   // sin((m+1)phi)
      cm = cn;
      sm = snn;
    }
  }

  __syncthreads();

  // ---------------- drain phase: LDS -> global ----------------
  const bool fullBlock = (blockStart + BLK) <= n;

  if (fullBlock) {
#if HAVE_TDM
    // One wave issues 11 Tensor Data Mover descriptors: each drains one
    // contiguous region (256*(2l+1) floats) LDS->global. TENSORcnt path;
    // EXEC is ignored by tensor ops, so gate to a single wave.
    if (tid < 32u) {
#pragma unroll
      for (int l = 0; l <= LMAX; ++l) {
        const unsigned w   = (unsigned)(2 * l + 1);
        const unsigned len = BLK * w;  // elements; <= 5376 (fits 16-bit tile_dim0)
        const unsigned ldsAddr = (unsigned)(uintptr_t)&tile[BLK * (unsigned)(l * l)];
        const unsigned long long ga =
            (unsigned long long)(uintptr_t)(out + (size_t)(l * l) * (size_t)n +
                                            (size_t)blockStart * w);
        // D# group 0: count=1 | lds_addr | global_addr[56:0] | type=2
        uint32x4 g0 = { 1u,
                        ldsAddr,
                        (unsigned)(ga & 0xFFFFFFFFu),
                        (unsigned)((ga >> 32) & 0x01FFFFFFu) | (2u << 30) };
        // D# group 1: data_size=2(4B); tensor_dim0=len; tensor_dim1=1;
        //             tile_dim0=len; tile_dim1=1; dim0_stride=len
        int32x8 g1 = { (int)0x00020000u,                 // data_size=2 (4 bytes)
                       (int)((len & 0xFFFFu) << 16),     // tensor_dim0[15:0]
                       (int)((len >> 16) | (1u << 16)),  // tensor_dim0[31:16] | tensor_dim1[15:0]=1
                       (int)((len & 0xFFFFu) << 16),     // tensor_dim1[31:16]=0 | tile_dim0
                       (int)1,                           // tile_dim1=1, tile_dim2=0
                       (int)len,                         // tensor_dim0_stride[31:0]
                       0,                                // stride0[47:32]=0 | stride1[15:0]=0
                       0 };                              // stride1[47:16]=0
        int32x4 gz  = { 0, 0, 0, 0 };
        int32x8 gz8 = { 0, 0, 0, 0, 0, 0, 0, 0 };  // unused VADDR4 slot (6-arg form)
        __builtin_amdgcn_tensor_store_from_lds(g0, g1, gz, gz, gz8, 0);
      }
      __builtin_amdgcn_s_wait_tensorcnt(0);
    }
#else
    // Fallback: CDNA5 async LDS->global b128 stores (ASYNCcnt), fully coalesced.
    const unsigned ldsBase = (unsigned)(uintptr_t)&tile[0];
#pragma unroll
    for (int l = 0; l <= LMAX; ++l) {
      const unsigned w = (unsigned)(2 * l + 1);
      const unsigned chunks = 64u * w;  // 16B chunks in region l
      const unsigned long long gbase =
          (unsigned long long)(uintptr_t)(out + (size_t)(l * l) * (size_t)n +
                                          (size_t)blockStart * w);
      const unsigned regionB = ldsBase + BLK * (unsigned)(l * l) * 4u;
      for (unsigned c = tid; c < chunks; c += BLK) {
        const unsigned goff = c * 16u;
        const unsigned loff = regionB + c * 16u;
        asm volatile("global_store_async_from_lds_b128 %0, %1, %2"
                     :: "v"(goff), "v"(loff), "s"(gbase) : "memory");
      }
    }
#endif
  } else {
    // Tail block: guarded scalar drain of valid rows only.
    const unsigned valid = n - blockStart;
#pragma unroll
    for (int l = 0; l <= LMAX; ++l) {
      const unsigned w = (unsigned)(2 * l + 1);
      const unsigned regionWords = BLK * (unsigned)(l * l);
      float* gdst = out + (size_t)(l * l) * (size_t)n + (size_t)blockStart * w;
      const unsigned elems = valid * w;
      for (unsigned e = tid; e < elems; e += BLK) {
        __builtin_nontemporal_store(tile[regionWords + e], gdst + e);
      }
    }
  }
  // s_endpgm performs an implicit wait-idle (TENSORcnt/ASYNCcnt -> 0) before LDS dealloc.
}

// ---------------- launcher ----------------
extern "C" void kernel_launch(void* const* d_in, const int* in_sizes, int n_in,
                              void* d_out, int out_size, void* d_ws, size_t ws_size,
                              hipStream_t stream) {
  const float* ct = (const float*)d_in[0];
  const float* ph = (const float*)d_in[1];
  float* out = (float*)d_out;
  const unsigned n = (unsigned)in_sizes[0];
  const unsigned grid = (n + BLK - 1) / BLK;
  hipLaunchKernelGGL(sph_harm_l10_kernel, dim3(grid), dim3(BLK), 0, stream,
                     ct, ph, out, n);
}